// STGCN_24790551233455
// MI455X (gfx1250) — compile-verified
//
#include <hip/hip_runtime.h>
#include <hip/hip_bf16.h>

// ---------------------------------------------------------------------------
// STGCN (4x GCNConv + linear head) for MI455X / gfx1250.
// Dense GEMMs use V_WMMA_F32_16X16X4_F32 (exact fp32, matrix pipe).
// Edge aggregation uses global_load_b128 + global_atomic_add_f32 (L2-resident).
// ---------------------------------------------------------------------------

#define N_TOTAL   60000
#define NUM_NODES 10000
#define N_EDGES   960000
#define IN_CH     32
#define HID       256
#define OUT_CH    12

typedef __attribute__((ext_vector_type(2))) float v2f;
typedef __attribute__((ext_vector_type(8))) float v8f;

static __device__ __forceinline__ v8f wmma_f32_16x16x4(v2f a, v2f b, v8f c) {
    // 8 args: (neg_a, A, neg_b, B, c_mod, C, reuse_a, reuse_b); f32: NEG[1:0]=0.
    return __builtin_amdgcn_wmma_f32_16x16x4_f32(false, a, false, b, (short)0, c,
                                                 false, false);
}

// ---------------------------------------------------------------------------
// C[M x 256] = A[M x K] * B[K x 256].  M % 16 == 0, K % 4 == 0.
// One wave -> 16x64 output strip (4 accumulators), so the A fragment is
// reused 4x per k-step. 256-thread blocks = 8 waves.
// ---------------------------------------------------------------------------
__global__ void gcn_gemm_wmma(const float* __restrict__ A,
                              const float* __restrict__ B,
                              float* __restrict__ C, int M, int K) {
    const int lane = threadIdx.x & 31;
    const int wave = blockIdx.x * (blockDim.x >> 5) + (threadIdx.x >> 5);

    const int waves_per_row = HID / 64;          // 4 strips of 64 cols
    const int tile_m = wave / waves_per_row;
    const int col_base = (wave % waves_per_row) * 64;
    if (tile_m * 16 >= M) return;                // wave-uniform: EXEC stays full

    const int mrow  = tile_m * 16 + (lane & 15); // A row for this lane
    const int khalf = (lane >> 4) << 1;          // lanes 16..31 hold K+2,K+3
    const int ncol  = lane & 15;                 // B/C column within 16-tile

    v8f acc0 = {}, acc1 = {}, acc2 = {}, acc3 = {};

    const float* __restrict__ arow = A + (long)mrow * K;
    for (int k0 = 0; k0 < K; k0 += 4) {
        v2f a;
        a.x = arow[k0 + khalf];
        a.y = arow[k0 + khalf + 1];

        const float* __restrict__ brow0 = B + (long)(k0 + khalf) * HID + col_base + ncol;
        const float* __restrict__ brow1 = brow0 + HID;

        v2f b0, b1, b2, b3;
        b0.x = brow0[0];  b0.y = brow1[0];
        b1.x = brow0[16]; b1.y = brow1[16];
        b2.x = brow0[32]; b2.y = brow1[32];
        b3.x = brow0[48]; b3.y = brow1[48];

        acc0 = wmma_f32_16x16x4(a, b0, acc0);
        acc1 = wmma_f32_16x16x4(a, b1, acc1);
        acc2 = wmma_f32_16x16x4(a, b2, acc2);
        acc3 = wmma_f32_16x16x4(a, b3, acc3);
    }

    // C/D layout: VGPR v holds row (v + 8*(lane>=16)), col = lane&15.
    const int rbase = tile_m * 16 + ((lane >> 4) << 3);
#pragma unroll
    for (int v = 0; v < 8; ++v) {
        float* crow = C + (long)(rbase + v) * HID + col_base + ncol;
        crow[0]  = acc0[v];
        crow[16] = acc1[v];
        crow[32] = acc2[v];
        crow[48] = acc3[v];
    }
}

// ---------------------------------------------------------------------------
// Degree / normalization precompute (shared by all 4 layers).
// ---------------------------------------------------------------------------
__global__ void deg_init(float* __restrict__ deg) {
    int i = blockIdx.x * blockDim.x + threadIdx.x;
    if (i < N_TOTAL) deg[i] = 1.0f;              // self-loop contribution
}

__global__ void deg_accum(const int* __restrict__ dst, float* __restrict__ deg) {
    int e = blockIdx.x * blockDim.x + threadIdx.x;
    if (e < N_EDGES) atomicAdd(&deg[dst[e]], 1.0f);
}

__global__ void deg_to_dinv(float* __restrict__ deg) {
    int i = blockIdx.x * blockDim.x + threadIdx.x;
    if (i < N_TOTAL) {
        float d = deg[i];
        deg[i] = (d > 0.0f) ? rsqrtf(d) : 0.0f;  // always >0 here (self-loops)
    }
}

__global__ void edge_norm(const int* __restrict__ src, const int* __restrict__ dst,
                          const float* __restrict__ dinv, float* __restrict__ norm) {
    int e = blockIdx.x * blockDim.x + threadIdx.x;
    if (e < N_EDGES) norm[e] = dinv[src[e]] * dinv[dst[e]];
}

// ---------------------------------------------------------------------------
// ACC[i, c] = bias[c]
// ---------------------------------------------------------------------------
__global__ void init_bias(float* __restrict__ acc, const float* __restrict__ bias) {
    int t = blockIdx.x * blockDim.x + threadIdx.x;   // blockDim == 256
    if (t < N_TOTAL * HID) acc[t] = bias[t & (HID - 1)];
}

// ---------------------------------------------------------------------------
// ACC[dst, :] += H[src, :] * norm[e].  64 threads/edge, float4 per thread.
// ---------------------------------------------------------------------------
__global__ void edge_scatter(const float* __restrict__ H,
                             const int* __restrict__ src, const int* __restrict__ dst,
                             const float* __restrict__ norm, float* __restrict__ acc) {
    int t = blockIdx.x * blockDim.x + threadIdx.x;
    int e = t >> 6;                                  // 64 threads per edge
    if (e >= N_EDGES) return;
    int c4 = (t & 63) << 2;

    int s = src[e];
    int d = dst[e];
    float w = norm[e];

    const float4 h4 = *reinterpret_cast<const float4*>(H + (long)s * HID + c4);
    float* ap = acc + (long)d * HID + c4;
    atomicAdd(ap + 0, h4.x * w);
    atomicAdd(ap + 1, h4.y * w);
    atomicAdd(ap + 2, h4.z * w);
    atomicAdd(ap + 3, h4.w * w);
}

// ---------------------------------------------------------------------------
// ACC[i, :] = relu(ACC[i, :] + H[i, :] * dinv[i]^2)    (self-loop + ReLU)
// ---------------------------------------------------------------------------
__global__ void selfloop_relu(const float* __restrict__ H,
                              const float* __restrict__ dinv,
                              float* __restrict__ acc) {
    int t = blockIdx.x * blockDim.x + threadIdx.x;
    int node = t >> 6;
    if (node >= N_TOTAL) return;
    int c4 = (t & 63) << 2;

    float di = dinv[node];
    float w = di * di;

    const float4 h4 = *reinterpret_cast<const float4*>(H + (long)node * HID + c4);
    float4* ap = reinterpret_cast<float4*>(acc + (long)node * HID + c4);
    float4 a = *ap;
    a.x = fmaxf(fmaf(h4.x, w, a.x), 0.0f);
    a.y = fmaxf(fmaf(h4.y, w, a.y), 0.0f);
    a.z = fmaxf(fmaf(h4.z, w, a.z), 0.0f);
    a.w = fmaxf(fmaf(h4.w, w, a.w), 0.0f);
    *ap = a;
}

// ---------------------------------------------------------------------------
// out[i, j] = sum_k ACC[50000 + i, k] * Wout[k, j] + bout[j]   (10000 x 12)
// Tiny (61 MFLOP) -> plain FMA kernel.
// ---------------------------------------------------------------------------
__global__ void head_gemm(const float* __restrict__ acc,
                          const float* __restrict__ Wout,
                          const float* __restrict__ bout,
                          float* __restrict__ out) {
    int t = blockIdx.x * blockDim.x + threadIdx.x;
    if (t >= NUM_NODES * OUT_CH) return;
    int i = t / OUT_CH;
    int j = t - i * OUT_CH;
    const float* hrow = acc + (long)(N_TOTAL - NUM_NODES + i) * HID;
    float sum = bout[j];
#pragma unroll 4
    for (int k = 0; k < HID; ++k)
        sum = fmaf(hrow[k], Wout[k * OUT_CH + j], sum);
    out[t] = sum;
}

// ---------------------------------------------------------------------------
extern "C" void kernel_launch(void* const* d_in, const int* in_sizes, int n_in,
                              void* d_out, int out_size, void* d_ws, size_t ws_size,
                              hipStream_t stream) {
    const float* x    = (const float*)d_in[0];
    const int*   ei   = (const int*)d_in[1];
    const float* W1   = (const float*)d_in[2];
    const float* b1   = (const float*)d_in[3];
    const float* W2   = (const float*)d_in[4];
    const float* b2   = (const float*)d_in[5];
    const float* W3   = (const float*)d_in[6];
    const float* b3   = (const float*)d_in[7];
    const float* W4   = (const float*)d_in[8];
    const float* b4   = (const float*)d_in[9];
    const float* Wout = (const float*)d_in[10];
    const float* bout = (const float*)d_in[11];
    float* out = (float*)d_out;

    const int* src = ei;
    const int* dst = ei + N_EDGES;

    // Workspace layout (16B-aligned slices):
    float* H    = (float*)d_ws;                        // 60000*256 f32
    float* ACC  = H   + (long)N_TOTAL * HID;           // 60000*256 f32
    float* dinv = ACC + (long)N_TOTAL * HID;           // 60000 f32 (deg in place)
    float* norm = dinv + N_TOTAL;                      // 960000 f32

    const int T = 256;
    const int gNode  = (N_TOTAL + T - 1) / T;
    const int gEdge  = (N_EDGES + T - 1) / T;
    const int gFeat  = (N_TOTAL * HID + T - 1) / T;    // 1 thread / element
    const int gEdge4 = (N_EDGES * 64 + T - 1) / T;     // 64 threads / edge
    const int gNode4 = (N_TOTAL * 64 + T - 1) / T;     // 64 threads / node
    const int gGemm  = ((N_TOTAL / 16) * 4 + 7) / 8;   // waves/8 per block
    const int gHead  = (NUM_NODES * OUT_CH + T - 1) / T;

    // --- normalization precompute (layer-invariant) ---
    deg_init   <<<gNode, T, 0, stream>>>(dinv);
    deg_accum  <<<gEdge, T, 0, stream>>>(dst, dinv);
    deg_to_dinv<<<gNode, T, 0, stream>>>(dinv);
    edge_norm  <<<gEdge, T, 0, stream>>>(src, dst, dinv, norm);

    // --- layer 1 (K = 32, input x) ---
    gcn_gemm_wmma<<<gGemm, T, 0, stream>>>(x, W1, H, N_TOTAL, IN_CH);
    init_bias    <<<gFeat, T, 0, stream>>>(ACC, b1);
    edge_scatter <<<gEdge4, T, 0, stream>>>(H, src, dst, norm, ACC);
    selfloop_relu<<<gNode4, T, 0, stream>>>(H, dinv, ACC);

    // --- layers 2..4 (K = 256, input ACC) ---
    const float* Ws[3] = {W2, W3, W4};
    const float* bs[3] = {b2, b3, b4};
    for (int l = 0; l < 3; ++l) {
        gcn_gemm_wmma<<<gGemm, T, 0, stream>>>(ACC, Ws[l], H, N_TOTAL, HID);
        init_bias    <<<gFeat, T, 0, stream>>>(ACC, bs[l]);
        edge_scatter <<<gEdge4, T, 0, stream>>>(H, src, dst, norm, ACC);
        selfloop_relu<<<gNode4, T, 0, stream>>>(H, dinv, ACC);
    }

    // --- projection head on the last 10000 nodes ---
    head_gemm<<<gHead, T, 0, stream>>>(ACC, Wout, bout, out);

    (void)in_sizes; (void)n_in; (void)out_size; (void)ws_size;
}